// FactorMask_14164802142736
// MI455X (gfx1250) — compile-verified
//
#include <hip/hip_runtime.h>
#include <stddef.h>
#include <stdint.h>

// Problem constants (match reference)
#define BATCH 8
#define HW    224
#define CH    32
#define NE    9
#define TILE_W 112           // output pixels per block (half a row)
#define LX     114           // tile pixels including 1-px halo each side
#define QPP    8             // float4 quads per pixel (CH/4)
#define NTHREADS 256

struct alignas(16) f4 { float v[4]; };
typedef float vfloat4 __attribute__((ext_vector_type(4)));

// ---- CDNA5 async global->LDS copy (GV mode: vdst = LDS byte addr, vaddr = 64b global) ----
__device__ __forceinline__ void async_ld_b128(uint32_t lds_addr, const float* gptr) {
    asm volatile("global_load_async_to_lds_b128 %0, %1, off"
                 :: "v"(lds_addr), "v"(gptr)
                 : "memory");
}
__device__ __forceinline__ void wait_async0() {
    asm volatile("s_wait_asynccnt 0x0" ::: "memory");
}

__global__ __launch_bounds__(NTHREADS) void factormask_kernel(
    const float* __restrict__ inp,     // [B,H,W,C]
    const float* __restrict__ kern,    // [E,1,1,1,C] -> flat [E*C]
    const float* __restrict__ mask,    // [E,1,1,1,C] -> flat [E*C]
    float* __restrict__ out)           // [B,H,W,C]
{
    __shared__ f4 tile[3 * LX * QPP];   // 3 rows x 114 px x 32ch = 43,776 B
    __shared__ f4 kq[NE * QPP];         // 1,152 B
    __shared__ f4 mq[NE * QPP];         // 1,152 B (holds |mask|, then scaled)
    __shared__ float red[NTHREADS];
    __shared__ float maxv;

    const int tid  = threadIdx.x;
    const int bx   = blockIdx.x;
    const int half = bx & 1;
    const int bh   = bx >> 1;
    const int h    = bh % HW;
    const int b    = bh / HW;
    const int w0   = half * TILE_W;
    const int wstart = w0 - 1;          // global w corresponding to lx = 0

    // ---- stage kernel/|mask| into LDS as b128; block-local max|mask| reduction ----
    float lm = 0.0f;
    if (tid < NE * QPP) {               // 72 quads cover all 288 values
        const f4 kk = ((const f4*)kern)[tid];
        const f4 mv = ((const f4*)mask)[tid];
        f4 ma;
        #pragma unroll
        for (int j = 0; j < 4; ++j) {
            ma.v[j] = fabsf(mv.v[j]);
            lm = fmaxf(lm, ma.v[j]);
        }
        kq[tid] = kk;
        mq[tid] = ma;
    }
    red[tid] = lm;
    __syncthreads();
    for (int s = NTHREADS / 2; s > 0; s >>= 1) {
        if (tid < s) red[tid] = fmaxf(red[tid], red[tid + s]);
        __syncthreads();
    }
    if (tid == 0) maxv = red[0];
    __syncthreads();
    const float minv = 1.0f / (maxv + 1e-6f);
    if (tid < NE * QPP) {
        f4 m = mq[tid];
        #pragma unroll
        for (int j = 0; j < 4; ++j) m.v[j] *= minv;
        mq[tid] = m;
    }

    // ---- zero-fill OOB halo; async-load the 113 valid pixel columns per row ----
    const int zlx  = half ? (LX - 1) : 0;   // the one out-of-image column
    const int lxlo = half ? 0 : 1;          // first valid column
    const f4 z4 = { {0.f, 0.f, 0.f, 0.f} };

    for (int r = 0; r < 3; ++r) {
        const int gr = h - 1 + r;
        if (gr < 0 || gr >= HW) {
            for (int q = tid; q < LX * QPP; q += NTHREADS)
                tile[r * LX * QPP + q] = z4;
        } else {
            if (tid < QPP) tile[(r * LX + zlx) * QPP + tid] = z4;
            const float* rowp = inp + ((size_t)(b * HW + gr) * HW) * CH;
            for (int q = tid; q < 113 * QPP; q += NTHREADS) {
                const int px = q >> 3;          // q / QPP
                const int cq = q & 7;           // q % QPP
                const int lx = lxlo + px;
                const int w  = wstart + lx;     // always in [0, 224)
                const uint32_t laddr =
                    (uint32_t)(uintptr_t)&tile[(r * LX + lx) * QPP + cq];
                async_ld_b128(laddr, rowp + (size_t)w * CH + cq * 4);
            }
        }
    }
    wait_async0();
    __syncthreads();

    // ---- compute: tap order e=0 center, then raster skipping center ----
    // e -> (dy, dx)
    const int dy[NE] = { 0, -1, -1, -1,  0,  0,  1,  1,  1 };
    const int dx[NE] = { 0, -1,  0,  1, -1,  1, -1,  0,  1 };
    const float invE = 1.0f / 9.0f;

    float* outp = out + (((size_t)b * HW + h) * HW + w0) * CH;

    for (int q = tid; q < TILE_W * QPP; q += NTHREADS) {
        const int px  = q >> 3;
        const int cq  = q & 7;
        const int lxc = px + 1;                 // center column in LDS

        f4 ad[NE];
        float n1[4] = {0.f, 0.f, 0.f, 0.f};
        float sd[4] = {0.f, 0.f, 0.f, 0.f};

        #pragma unroll
        for (int e = 0; e < NE; ++e) {
            const f4 t  = tile[((dy[e] + 1) * LX + (lxc + dx[e])) * QPP + cq];
            const f4 mm = mq[e * QPP + cq];
            const f4 kk = kq[e * QPP + cq];
            #pragma unroll
            for (int j = 0; j < 4; ++j) {
                const float d = fmaf(t.v[j], mm.v[j], -kk.v[j]);
                sd[j] += d;
                const float a = fabsf(d);
                ad[e].v[j] = a;
                n1[j] += a;
            }
        }

        float var[4] = {0.f, 0.f, 0.f, 0.f};
        #pragma unroll
        for (int j = 0; j < 4; ++j) {
            const float mean = sd[j] * invE;
            #pragma unroll
            for (int e = 0; e < NE; ++e)
                var[j] += fabsf(ad[e].v[j] - mean);
        }

        vfloat4 o;
        #pragma unroll
        for (int j = 0; j < 4; ++j) {
            const float t1 = fmaf(var[j], -invE, 1.0f);
            const float t2 = fmaf(n1[j],  -invE, 1.0f);
            o[j] = t1 * t2;
        }
        // Output is write-once / never re-read: non-temporal store keeps the
        // 192 MB L2 free for input rows (re-used 3x by neighboring blocks).
        __builtin_nontemporal_store(o, (vfloat4*)(outp + (size_t)px * CH + cq * 4));
    }
}

extern "C" void kernel_launch(void* const* d_in, const int* in_sizes, int n_in,
                              void* d_out, int out_size, void* d_ws, size_t ws_size,
                              hipStream_t stream) {
    (void)in_sizes; (void)n_in; (void)out_size; (void)d_ws; (void)ws_size;
    const float* inp  = (const float*)d_in[0];
    const float* kern = (const float*)d_in[1];
    const float* mask = (const float*)d_in[2];
    float* out = (float*)d_out;

    const int nblocks = BATCH * HW * 2;   // one block per (b, h, half-row)
    factormask_kernel<<<nblocks, NTHREADS, 0, stream>>>(inp, kern, mask, out);
}